// PointNetSetAbstractionLite_27547920236659
// MI455X (gfx1250) — compile-verified
//
#include <hip/hip_runtime.h>

// PointNet Set Abstraction (lite), fused for gfx1250 (MI455X).
// B=8, N=16384, C=64, S=2048, K=32, MLP 67->128->128, maxpool over K.
// f32 WMMA (V_WMMA_F32_16X16X4_F32). N-blocked halves keep 8 v8f accs live;
// launch_bounds(256,1) + pair-interleaved LDS weights kill spills/moves.

#define B_   8
#define N_   16384
#define C_   64
#define S_   2048
#define K_   32
#define OC_  128
#define INPAD 68     // 3 rel_xyz + 1 zero pad column + 64 feat  (17 K-steps of 4)
#define RPS  288     // weight pair-row stride (floats): 288%64==32, base 32-aligned
                     //  -> half-wave B reads hit disjoint 32-bank spans (conflict-free)
#define W1P  34      // INPAD/2 pair rows
#define W2P  64      // OC_/2  pair rows
#define ILD  68      // In row stride:  68%64==4  -> 4-bank rotation (conflict-free A)
#define SLD  132     // X1 row stride: 132%64==4  -> 4-bank rotation (conflict-free A)
#define TILE_ 512

typedef __attribute__((ext_vector_type(2))) float v2f;
typedef __attribute__((ext_vector_type(8))) float v8f;

__device__ __forceinline__ v8f wmma4(v2f a, v2f b, v8f c) {
  // (neg_a, A, neg_b, B, c_mod, C, reuse_a, reuse_b)
  return __builtin_amdgcn_wmma_f32_16x16x4_f32(false, a, false, b, (short)0, c,
                                               false, false);
}

// ---------------------------------------------------------------------------
// Kernel 1: deterministic per-batch sampling (odd-multiplier bijection mod N)
// + gather of new_xyz straight into d_out.
// ---------------------------------------------------------------------------
__global__ void sa_sample(const float* __restrict__ xyz,
                          const int* __restrict__ seedp,
                          float* __restrict__ new_xyz) {
  int t = blockIdx.x * 256 + threadIdx.x;        // [0, B_*S_)
  int b = t >> 11;
  int s = t & (S_ - 1);
  unsigned h = (unsigned)seedp[0] * 0x9E3779B9u ^ (unsigned)(b * 17) * 0x85EBCA6Bu;
  h ^= h >> 16; h *= 0x7FEB352Du; h ^= h >> 15; h *= 0x846CA68Bu; h ^= h >> 16;
  unsigned a = (h << 1) | 1u;                    // odd -> bijection mod 2^14
  int n = (int)((a * (unsigned)s + (h >> 5)) & (unsigned)(N_ - 1));
  const float* p = xyz + ((size_t)b * N_ + n) * 3;
  new_xyz[(size_t)t * 3 + 0] = p[0];
  new_xyz[(size_t)t * 3 + 1] = p[1];
  new_xyz[(size_t)t * 3 + 2] = p[2];
}

// ---------------------------------------------------------------------------
// Kernel 2: kNN, one thread per query. Point tiles staged in LDS (broadcast
// reads). Per-thread top-32 list in LDS [slot][tid] (lane->fixed bank).
// ---------------------------------------------------------------------------
__global__ void sa_knn(const float* __restrict__ xyz,
                       const float* __restrict__ new_xyz,
                       int* __restrict__ knn) {
  extern __shared__ float smem[];
  float* ptile = smem;                        // [TILE_][4]
  float* kd    = ptile + TILE_ * 4;           // [K_][256]
  int*   ki    = (int*)(kd + K_ * 256);       // [K_][256]

  const int tid = threadIdx.x;                // 0..255
  const int q   = blockIdx.x * 256 + tid;     // queries of one batch per block
  const int b   = q >> 11;
  const float qx = new_xyz[(size_t)q * 3 + 0];
  const float qy = new_xyz[(size_t)q * 3 + 1];
  const float qz = new_xyz[(size_t)q * 3 + 2];

#pragma unroll
  for (int k = 0; k < K_; ++k) { kd[k * 256 + tid] = 3.4e38f; ki[k * 256 + tid] = 0; }
  float curmax = 3.4e38f;
  int   curarg = 0;

  const float* base = xyz + (size_t)b * N_ * 3;
  for (int t0 = 0; t0 < N_; t0 += TILE_) {
    __syncthreads();                           // previous tile fully consumed
    for (int i = tid; i < TILE_; i += 256) {
      const float* p = base + (size_t)(t0 + i) * 3;
      ptile[i * 4 + 0] = p[0];
      ptile[i * 4 + 1] = p[1];
      ptile[i * 4 + 2] = p[2];
    }
    __syncthreads();
    for (int p = 0; p < TILE_; ++p) {          // uniform address -> LDS broadcast
      float dx = qx - ptile[p * 4 + 0];
      float dy = qy - ptile[p * 4 + 1];
      float dz = qz - ptile[p * 4 + 2];
      float d2 = dx * dx + dy * dy + dz * dz;
      if (d2 < curmax) {
        kd[curarg * 256 + tid] = d2;
        ki[curarg * 256 + tid] = t0 + p;
        float m = kd[0 * 256 + tid]; int a = 0;
#pragma unroll
        for (int k = 1; k < K_; ++k) {
          float v = kd[k * 256 + tid];
          if (v > m) { m = v; a = k; }
        }
        curmax = m; curarg = a;
      }
    }
  }
#pragma unroll
  for (int k = 0; k < K_; ++k) knn[(size_t)q * K_ + k] = ki[k * 256 + tid];
}

// ---------------------------------------------------------------------------
// Kernel 3: fused gather -> MLP1 -> MLP2 -> maxpool, one query per wave.
// Weights stored K-pair-interleaved in LDS: pair row p holds (W[2p][c],W[2p+1][c])
// adjacent, so each B operand is a single aligned ds_load_b64.
// ---------------------------------------------------------------------------
__global__ void __launch_bounds__(256, 1)
sa_mlp(const float* __restrict__ xyz,
       const float* __restrict__ feat,
       const float* __restrict__ W1, const float* __restrict__ b1,
       const float* __restrict__ W2, const float* __restrict__ b2,
       const int* __restrict__ knn,
       const float* __restrict__ new_xyz,
       float* __restrict__ new_feat) {
  extern __shared__ float smem[];
  float* W1s  = smem;                     // [W1P][RPS] pair-interleaved, zero row at r==3
  float* W2s  = W1s + W1P * RPS;          // [W2P][RPS] pair-interleaved
  float* b1s  = W2s + W2P * RPS;          // [OC_]
  float* b2s  = b1s + OC_;                // [OC_]
  float* scr0 = b2s + OC_;                // 8 waves * ( In[K_][ILD] + X1[K_][SLD] )

  const int tid  = threadIdx.x;
  const int lane = tid & 31;
  const int wave = tid >> 5;
  float* inb = scr0 + wave * (K_ * ILD + K_ * SLD);   // In  [32][68]
  float* x1b = inb + K_ * ILD;                        // X1  [32][132]

  // Stage weights+bias into LDS (pair-interleaved). W1 gets a zero row at
  // index 3 so the input concat [rel_xyz(3), 0, feat(64)] pads K to 68.
  for (int i = tid; i < INPAD * OC_; i += 256) {
    int r = i >> 7, c = i & (OC_ - 1);
    float v = (r < 3) ? W1[r * OC_ + c] : ((r == 3) ? 0.f : W1[(r - 1) * OC_ + c]);
    W1s[(r >> 1) * RPS + 2 * c + (r & 1)] = v;
  }
  for (int i = tid; i < OC_ * OC_; i += 256) {
    int r = i >> 7, c = i & (OC_ - 1);
    W2s[(r >> 1) * RPS + 2 * c + (r & 1)] = W2[i];
  }
  if (tid < OC_) { b1s[tid] = b1[tid]; b2s[tid] = b2[tid]; }
  __syncthreads();

  const int l15  = lane & 15;
  const int koff = (lane >> 4) * 2;       // A split-lane K offset (0 or 2)
  const int kofp = (lane >> 4);           // B split-lane K pair offset (0 or 1)
  const int rhalf = (lane >> 4) * 8;      // C-layout row offset for upper lanes
  const v8f vzero = {0.f, 0.f, 0.f, 0.f, 0.f, 0.f, 0.f, 0.f};
  const int nwaves = gridDim.x * 8;

  for (int q = blockIdx.x * 8 + wave; q < B_ * S_; q += nwaves) {
    const int b = q >> 11;

    // ---- gather: lane == neighbor; build In[32][68] ------------------------
    {
      int n = knn[(size_t)q * K_ + lane];
      float qx = new_xyz[(size_t)q * 3 + 0];
      float qy = new_xyz[(size_t)q * 3 + 1];
      float qz = new_xyz[(size_t)q * 3 + 2];
      const float* pr = xyz + ((size_t)b * N_ + n) * 3;
      float* row = inb + lane * ILD;
      row[0] = pr[0] - qx; row[1] = pr[1] - qy; row[2] = pr[2] - qz; row[3] = 0.f;
      const float* fr = feat + ((size_t)b * N_ + n) * C_;
#pragma unroll
      for (int c = 0; c < C_; c += 4) {
        float4 v = *(const float4*)(fr + c);
        *(float4*)(row + 4 + c) = v;
      }
    }

    // ---- layer 1: X1 = relu(In @ W1 + b1), 32x68x128, two 64-col halves ----
#pragma unroll 1
    for (int h = 0; h < 2; ++h) {
      v8f acc[2][4];
#pragma unroll
      for (int mt = 0; mt < 2; ++mt)
#pragma unroll
        for (int nt = 0; nt < 4; ++nt) acc[mt][nt] = vzero;

#pragma unroll 2
      for (int k = 0; k < INPAD; k += 4) {
        v2f a0 = *(const v2f*)(inb + l15 * ILD + k + koff);
        v2f a1 = *(const v2f*)(inb + (l15 + 16) * ILD + k + koff);
        const float* wrow = W1s + (k / 2 + kofp) * RPS;
#pragma unroll
        for (int nt = 0; nt < 4; ++nt) {
          int col = h * 64 + nt * 16 + l15;
          v2f bb = *(const v2f*)(wrow + 2 * col);
          acc[0][nt] = wmma4(a0, bb, acc[0][nt]);
          acc[1][nt] = wmma4(a1, bb, acc[1][nt]);
        }
      }
      // bias + relu -> X1 buffer (separate from In; no WAR with other half)
#pragma unroll
      for (int mt = 0; mt < 2; ++mt)
#pragma unroll
        for (int nt = 0; nt < 4; ++nt) {
          int col = h * 64 + nt * 16 + l15;
          float bv = b1s[col];
#pragma unroll
          for (int r = 0; r < 8; ++r) {
            float v = fmaxf(acc[mt][nt][r] + bv, 0.f);
            x1b[(mt * 16 + r + rhalf) * SLD + col] = v;
          }
        }
    }

    // ---- layer 2 + maxpool: relu(X1 @ W2 + b2), two 64-col halves ----------
#pragma unroll 1
    for (int h = 0; h < 2; ++h) {
      v8f acc[2][4];
#pragma unroll
      for (int mt = 0; mt < 2; ++mt)
#pragma unroll
        for (int nt = 0; nt < 4; ++nt) acc[mt][nt] = vzero;

#pragma unroll 2
      for (int k = 0; k < OC_; k += 4) {
        v2f a0 = *(const v2f*)(x1b + l15 * SLD + k + koff);
        v2f a1 = *(const v2f*)(x1b + (l15 + 16) * SLD + k + koff);
        const float* wrow = W2s + (k / 2 + kofp) * RPS;
#pragma unroll
        for (int nt = 0; nt < 4; ++nt) {
          int col = h * 64 + nt * 16 + l15;
          v2f bb = *(const v2f*)(wrow + 2 * col);
          acc[0][nt] = wmma4(a0, bb, acc[0][nt]);
          acc[1][nt] = wmma4(a1, bb, acc[1][nt]);
        }
      }
      // bias + relu + maxpool over the 32 neighbor rows, store this half
#pragma unroll
      for (int nt = 0; nt < 4; ++nt) {
        int col = h * 64 + nt * 16 + l15;
        float bv = b2s[col];
        float m = 0.f;                       // relu output >= 0
#pragma unroll
        for (int mt = 0; mt < 2; ++mt)
#pragma unroll
          for (int r = 0; r < 8; ++r) {
            float v = fmaxf(acc[mt][nt][r] + bv, 0.f);
            m = fmaxf(m, v);
          }
        m = fmaxf(m, __shfl_xor(m, 16));     // merge rows held by lane l and l+16
        if (lane < 16) new_feat[(size_t)q * OC_ + h * 64 + nt * 16 + lane] = m;
      }
    }
  }
}

// ---------------------------------------------------------------------------
extern "C" void kernel_launch(void* const* d_in, const int* in_sizes, int n_in,
                              void* d_out, int out_size, void* d_ws, size_t ws_size,
                              hipStream_t stream) {
  (void)in_sizes; (void)n_in; (void)out_size; (void)ws_size;
  const float* xyz  = (const float*)d_in[0];
  const float* feat = (const float*)d_in[1];
  const float* W1   = (const float*)d_in[2];
  const float* b1   = (const float*)d_in[3];
  const float* W2   = (const float*)d_in[4];
  const float* b2   = (const float*)d_in[5];
  const int*   seed = (const int*)d_in[6];

  float* out      = (float*)d_out;
  float* new_xyz  = out;                        // (B,S,3)
  float* new_feat = out + (size_t)B_ * S_ * 3;  // (B,S,128)
  int*   knn      = (int*)d_ws;                 // B*S*K ints (2 MB)

  size_t knn_smem = (size_t)(TILE_ * 4 + K_ * 256) * sizeof(float)
                  + (size_t)K_ * 256 * sizeof(int);                 // ~72 KB
  size_t mlp_smem = (size_t)(W1P * RPS + W2P * RPS + 2 * OC_
                             + 8 * (K_ * ILD + K_ * SLD)) * sizeof(float); // 318,720 B

  hipFuncSetAttribute((const void*)sa_knn,
                      hipFuncAttributeMaxDynamicSharedMemorySize, (int)knn_smem);
  hipFuncSetAttribute((const void*)sa_mlp,
                      hipFuncAttributeMaxDynamicSharedMemorySize, (int)mlp_smem);

  sa_sample<<<(B_ * S_) / 256, 256, 0, stream>>>(xyz, seed, new_xyz);
  sa_knn<<<(B_ * S_) / 256, 256, knn_smem, stream>>>(xyz, new_xyz, knn);
  sa_mlp<<<512, 256, mlp_smem, stream>>>(xyz, feat, W1, b1, W2, b2, knn,
                                         new_xyz, new_feat);
}